// RoiAlign_20607253086644
// MI455X (gfx1250) — compile-verified
//
#include <hip/hip_runtime.h>
#include <math.h>
#include <stdint.h>

// ---------------------------------------------------------------------------
// RoiAlign-style pipeline for MI455X (gfx1250, wave32).
//   stage 1: scores = rowmax(cls[100000,80]); keys = order-preserving u32
//   stage 2: 4-pass 8-bit radix select -> exact K-th threshold + tie handling
//   stage 3: one-block rank sorts (score desc/idx asc, then stable by level)
//   stage 4: cls gather; boxes + normalized boxes written
//   stage 5: crop_and_resize 500 x 14x14x256 via async global->LDS (B128)
// Dominant cost: stage 5, ~100MB stores + L2-resident gathered reads
// (~10-20us roofline at 23.3 TB/s). No matmul structure -> WMMA N/A; the
// CDNA5 path exercised is the ASYNCcnt-tracked async-copy engine.
// ---------------------------------------------------------------------------

#define K_TOP 500
#define NCLS  80
#define CCH   256

#define CTL_PREFIX 0
#define CTL_ABOVE  1
#define CTL_CNTGT  2
#define CTL_CNTEQ  3
#define CTL_HIST   16

#define AS1 __attribute__((address_space(1)))
#define AS3 __attribute__((address_space(3)))

typedef int v4i __attribute__((ext_vector_type(4)));

#if __has_builtin(__builtin_amdgcn_global_load_async_to_lds_b128)
#define HAVE_ASYNC128 1
#endif
#if __has_builtin(__builtin_amdgcn_global_load_async_to_lds_b32)
#define HAVE_ASYNC 1
#endif

#if __has_builtin(__builtin_amdgcn_s_wait_asynccnt)
#define WAIT_ASYNC(n) __builtin_amdgcn_s_wait_asynccnt(n)
#else
#define WAIT_ASYNC(n) asm volatile("s_wait_asynccnt " #n ::: "memory")
#endif

#ifdef HAVE_ASYNC128
__device__ __forceinline__ void async_copy_b128(const float* g, float* l) {
  void* gnc = const_cast<void*>((const void*)g);
  __builtin_amdgcn_global_load_async_to_lds_b128(
      (AS1 v4i*)gnc, (AS3 v4i*)l, 0, 0);
}
#endif
#ifdef HAVE_ASYNC
__device__ __forceinline__ void async_copy_f32(const float* g, float* l) {
  void* gnc = const_cast<void*>((const void*)g);
  __builtin_amdgcn_global_load_async_to_lds_b32(
      (AS1 int*)gnc, (AS3 int*)l, 0, 0);
}
#endif

// ---------------------------------------------------------------------------
__global__ void k_init(uint32_t* ctl) {
  int t = threadIdx.x;
  if (t < 16) ctl[t] = 0u;
  ctl[CTL_HIST + t] = 0u;  // 256 threads
}

__global__ void k_scores(const float* __restrict__ cls,
                         uint32_t* __restrict__ keys, int N) {
  int i = blockIdx.x * blockDim.x + threadIdx.x;
  if (i >= N) return;
  const float4* p = (const float4*)(cls + (size_t)i * NCLS);
  float m = -INFINITY;
#pragma unroll
  for (int k = 0; k < NCLS / 4; ++k) {
    float4 v = p[k];
    m = fmaxf(m, fmaxf(fmaxf(v.x, v.y), fmaxf(v.z, v.w)));
  }
  uint32_t u = __float_as_uint(m);
  keys[i] = (u & 0x80000000u) ? ~u : (u | 0x80000000u);
}

__global__ void k_hist(const uint32_t* __restrict__ keys, uint32_t* ctl,
                       int N, int shift, uint32_t maskHi) {
  __shared__ uint32_t lh[256];
  lh[threadIdx.x] = 0u;
  __syncthreads();
  uint32_t prefix = ctl[CTL_PREFIX];
  int i = blockIdx.x * blockDim.x + threadIdx.x;
  if (i < N) {
    uint32_t k = keys[i];
    if ((k & maskHi) == prefix) atomicAdd(&lh[(k >> shift) & 255u], 1u);
  }
  __syncthreads();
  uint32_t h = lh[threadIdx.x];
  if (h) atomicAdd(&ctl[CTL_HIST + threadIdx.x], h);
}

__global__ void k_select(uint32_t* ctl, int shift) {
  if (threadIdx.x == 0) {
    uint32_t cum = ctl[CTL_ABOVE];
    uint32_t pref = ctl[CTL_PREFIX];
    for (int d = 255; d >= 0; --d) {
      uint32_t h = ctl[CTL_HIST + d];
      if (cum + h >= (uint32_t)K_TOP) {
        ctl[CTL_PREFIX] = pref | ((uint32_t)d << shift);
        ctl[CTL_ABOVE] = cum;
        break;
      }
      cum += h;
    }
  }
  __syncthreads();
  ctl[CTL_HIST + threadIdx.x] = 0u;  // reset for next pass
}

__global__ void k_compact(const uint32_t* __restrict__ keys, uint32_t* ctl,
                          uint32_t* gtIdx, uint32_t* tieIdx, int N) {
  int i = blockIdx.x * blockDim.x + threadIdx.x;
  if (i >= N) return;
  uint32_t T = ctl[CTL_PREFIX];
  uint32_t k = keys[i];
  if (k > T) {
    uint32_t p = atomicAdd(&ctl[CTL_CNTGT], 1u);
    if (p < K_TOP) gtIdx[p] = (uint32_t)i;
  } else if (k == T) {
    uint32_t p = atomicAdd(&ctl[CTL_CNTEQ], 1u);
    if (p < K_TOP) tieIdx[p] = (uint32_t)i;
  }
}

__global__ void k_finalize(const uint32_t* __restrict__ keys,
                           const uint32_t* __restrict__ gtIdx,
                           const uint32_t* __restrict__ tieIdx,
                           const uint32_t* __restrict__ ctl,
                           const float* __restrict__ boxes,
                           const int* __restrict__ image_shape,
                           uint32_t* __restrict__ perm, int* __restrict__ lvlOut,
                           float* __restrict__ nb, float* __restrict__ out_b) {
  __shared__ uint32_t sIdx[K_TOP];
  __shared__ uint32_t sKey[K_TOP];
  __shared__ uint32_t sSorted[K_TOP];
  __shared__ int sLvl[K_TOP];
  __shared__ uint32_t sTie[K_TOP];

  int tid = threadIdx.x;
  int bdim = blockDim.x;
  uint32_t nGT = ctl[CTL_CNTGT]; if (nGT > K_TOP) nGT = K_TOP;
  uint32_t nEQ = ctl[CTL_CNTEQ]; if (nEQ > K_TOP) nEQ = K_TOP;
  uint32_t need = K_TOP - nGT;

  for (uint32_t i = tid; i < nGT; i += bdim) sIdx[i] = gtIdx[i];
  for (uint32_t i = tid; i < nEQ; i += bdim) sTie[i] = tieIdx[i];
  __syncthreads();
  for (uint32_t i = tid; i < nEQ; i += bdim) {
    uint32_t v = sTie[i], r = 0;
    for (uint32_t j = 0; j < nEQ; ++j) r += (sTie[j] < v) ? 1u : 0u;
    if (r < need) sIdx[nGT + r] = v;
  }
  __syncthreads();
  for (int i = tid; i < K_TOP; i += bdim) sKey[i] = keys[sIdx[i]];
  __syncthreads();
  for (int i = tid; i < K_TOP; i += bdim) {
    uint32_t k = sKey[i], id = sIdx[i];
    int r = 0;
    for (int j = 0; j < K_TOP; ++j) {
      uint32_t kj = sKey[j];
      r += ((kj > k) || (kj == k && sIdx[j] < id)) ? 1 : 0;
    }
    sSorted[r] = id;
  }
  __syncthreads();
  for (int i = tid; i < K_TOP; i += bdim) {
    uint32_t b = sSorted[i];
    float x1 = boxes[b * 4 + 0], y1 = boxes[b * 4 + 1];
    float x2 = boxes[b * 4 + 2], y2 = boxes[b * 4 + 3];
    float size = sqrtf((x2 - x1) * (y2 - y1));
    float lf = floorf(1.0f + log2f(size / 224.0f + 1e-7f));
    lf = fminf(fmaxf(lf, 0.0f), 4.0f);
    sLvl[i] = (int)lf;
  }
  __syncthreads();
  float Hi = (float)image_shape[1];
  float Wi = (float)image_shape[2];
  for (int i = tid; i < K_TOP; i += bdim) {
    int l = sLvl[i];
    int r = 0;
    for (int j = 0; j < K_TOP; ++j) {
      int lj = sLvl[j];
      r += ((lj < l) || (lj == l && j < i)) ? 1 : 0;
    }
    uint32_t b = sSorted[i];
    perm[r] = b;
    lvlOut[r] = l;
    float x1 = boxes[b * 4 + 0], y1 = boxes[b * 4 + 1];
    float x2 = boxes[b * 4 + 2], y2 = boxes[b * 4 + 3];
    out_b[r * 4 + 0] = x1; out_b[r * 4 + 1] = y1;
    out_b[r * 4 + 2] = x2; out_b[r * 4 + 3] = y2;
    nb[r * 4 + 0] = y1 / Hi; nb[r * 4 + 1] = x1 / Wi;
    nb[r * 4 + 2] = y2 / Hi; nb[r * 4 + 3] = x2 / Wi;
  }
}

__global__ void k_cls(const float* __restrict__ cls,
                      const uint32_t* __restrict__ perm,
                      float* __restrict__ out_cls) {
  int g = blockIdx.x * blockDim.x + threadIdx.x;
  if (g >= K_TOP * NCLS) return;
  int s = g / NCLS, c = g - s * NCLS;
  out_cls[g] = cls[(size_t)perm[s] * NCLS + c];
}

__device__ __forceinline__ void roi_xcoords(float x1n, float x2n, float Wf,
                                            int W, int tx, int& x0i, int& x1i,
                                            float& wx, bool& vx) {
  float in_x = x1n * Wf + (((float)tx) / 13.0f) * (x2n - x1n) * Wf;
  float x0f = floorf(in_x);
  wx = in_x - x0f;
  int xi = (int)x0f;
  x0i = xi < 0 ? 0 : (xi > W - 1 ? W - 1 : xi);
  x1i = (x0i + 1 > W - 1) ? W - 1 : x0i + 1;
  vx = (in_x >= 0.0f) && (in_x <= Wf);
}

// grid = (K_TOP, 14); block = 256 (one lane per channel, 8 wave32 waves)
__global__ __launch_bounds__(256) void k_roi(
    const float* __restrict__ f0, const float* __restrict__ f1,
    const float* __restrict__ f2, const float* __restrict__ f3,
    const float* __restrict__ f4, const int* __restrict__ lvl,
    const float* __restrict__ nb, float* __restrict__ out_roi) {
  const int s = blockIdx.x, ty = blockIdx.y, c = threadIdx.x;
  int l = lvl[s];
  int H, W;
  const float* F;
  switch (l) {
    case 0: H = 100; W = 160; F = f0; break;
    case 1: H = 50;  W = 80;  F = f1; break;
    case 2: H = 25;  W = 40;  F = f2; break;
    case 3: H = 13;  W = 20;  F = f3; break;
    default: H = 7;  W = 10;  F = f4; break;
  }
  float y1n = nb[s * 4 + 0], x1n = nb[s * 4 + 1];
  float y2n = nb[s * 4 + 2], x2n = nb[s * 4 + 3];
  float Hf = (float)(H - 1), Wf = (float)(W - 1);

  float in_y = y1n * Hf + (((float)ty) / 13.0f) * (y2n - y1n) * Hf;
  float y0f = floorf(in_y);
  float wy = in_y - y0f;
  int yi = (int)y0f;
  int y0i = yi < 0 ? 0 : (yi > H - 1 ? H - 1 : yi);
  int y1i = (y0i + 1 > H - 1) ? H - 1 : y0i + 1;
  bool vy = (in_y >= 0.0f) && (in_y <= Hf);

  float* orow = out_roi + (((size_t)s * 14 + ty) * 14) * CCH + c;

#if defined(HAVE_ASYNC128)
  // CDNA5 path, B128 grain: lane t moves 4 contiguous channels of tap t>>6
  // (one async b128 per lane per tx; 512B contiguous per wave per burst).
  // Consume crosses lanes -> barrier after wait; second barrier protects the
  // double buffer from being re-issued while still being read.
  __shared__ float abuf[2][4][CCH];
  const int tap = c >> 6;
  const int ch4 = (c & 63) << 2;
  const size_t taprow = ((tap < 2) ? (size_t)y0i : (size_t)y1i) * W;
  auto issue128 = [&](int tx, int bi) {
    int x0i, x1i; float wx; bool vx;
    roi_xcoords(x1n, x2n, Wf, W, tx, x0i, x1i, wx, vx);
    int xcol = (tap & 1) ? x1i : x0i;
    const float* src = F + (taprow + xcol) * CCH + ch4;
    async_copy_b128(src, &abuf[bi][tap][ch4]);
  };
  issue128(0, 0);
  int cur = 0;
#pragma unroll
  for (int tx = 0; tx < 14; ++tx) {
    if (tx < 13) {
      issue128(tx + 1, cur ^ 1);
      WAIT_ASYNC(1);
    } else {
      WAIT_ASYNC(0);
    }
    __syncthreads();  // all waves' asyncs for abuf[cur] have landed
    int x0i, x1i; float wx; bool vx;
    roi_xcoords(x1n, x2n, Wf, W, tx, x0i, x1i, wx, vx);
    volatile const float* vb = (volatile const float*)&abuf[cur][0][0];
    float tl = vb[0 * CCH + c], tr = vb[1 * CCH + c];
    float bl = vb[2 * CCH + c], br = vb[3 * CCH + c];
    float top = tl + (tr - tl) * wx;
    float bot = bl + (br - bl) * wx;
    float val = top + (bot - top) * wy;
    orow[(size_t)tx * CCH] = (vy && vx) ? val : 0.0f;
    __syncthreads();  // abuf[cur] fully consumed before next re-issue
    cur ^= 1;
  }
#elif defined(HAVE_ASYNC)
  // B32 grain: each lane stages its own 4 taps; same-lane consume, no barriers.
  size_t row0 = (size_t)y0i * W * CCH + c;
  size_t row1 = (size_t)y1i * W * CCH + c;
  __shared__ float abuf[2][4][CCH];
  {
    int x0i, x1i; float wx; bool vx;
    roi_xcoords(x1n, x2n, Wf, W, 0, x0i, x1i, wx, vx);
    async_copy_f32(F + row0 + (size_t)x0i * CCH, &abuf[0][0][c]);
    async_copy_f32(F + row0 + (size_t)x1i * CCH, &abuf[0][1][c]);
    async_copy_f32(F + row1 + (size_t)x0i * CCH, &abuf[0][2][c]);
    async_copy_f32(F + row1 + (size_t)x1i * CCH, &abuf[0][3][c]);
  }
  int cur = 0;
#pragma unroll
  for (int tx = 0; tx < 14; ++tx) {
    if (tx < 13) {
      int x0i, x1i; float wx; bool vx;
      roi_xcoords(x1n, x2n, Wf, W, tx + 1, x0i, x1i, wx, vx);
      int nb_ = cur ^ 1;
      async_copy_f32(F + row0 + (size_t)x0i * CCH, &abuf[nb_][0][c]);
      async_copy_f32(F + row0 + (size_t)x1i * CCH, &abuf[nb_][1][c]);
      async_copy_f32(F + row1 + (size_t)x0i * CCH, &abuf[nb_][2][c]);
      async_copy_f32(F + row1 + (size_t)x1i * CCH, &abuf[nb_][3][c]);
      WAIT_ASYNC(4);
    } else {
      WAIT_ASYNC(0);
    }
    int x0i, x1i; float wx; bool vx;
    roi_xcoords(x1n, x2n, Wf, W, tx, x0i, x1i, wx, vx);
    volatile const float* vb = (volatile const float*)&abuf[cur][0][0];
    float tl = vb[0 * CCH + c], tr = vb[1 * CCH + c];
    float bl = vb[2 * CCH + c], br = vb[3 * CCH + c];
    float top = tl + (tr - tl) * wx;
    float bot = bl + (br - bl) * wx;
    float val = top + (bot - top) * wy;
    orow[(size_t)tx * CCH] = (vy && vx) ? val : 0.0f;
    cur ^= 1;
  }
#else
  size_t row0 = (size_t)y0i * W * CCH + c;
  size_t row1 = (size_t)y1i * W * CCH + c;
#pragma unroll
  for (int tx = 0; tx < 14; ++tx) {
    int x0i, x1i; float wx; bool vx;
    roi_xcoords(x1n, x2n, Wf, W, tx, x0i, x1i, wx, vx);
    float tl = F[row0 + (size_t)x0i * CCH];
    float tr = F[row0 + (size_t)x1i * CCH];
    float bl = F[row1 + (size_t)x0i * CCH];
    float br = F[row1 + (size_t)x1i * CCH];
    float top = tl + (tr - tl) * wx;
    float bot = bl + (br - bl) * wx;
    float val = top + (bot - top) * wy;
    orow[(size_t)tx * CCH] = (vy && vx) ? val : 0.0f;
  }
#endif
}

// ---------------------------------------------------------------------------
extern "C" void kernel_launch(void* const* d_in, const int* in_sizes, int n_in,
                              void* d_out, int out_size, void* d_ws,
                              size_t ws_size, hipStream_t stream) {
  const int* image_shape = (const int*)d_in[0];
  const float* boxes = (const float*)d_in[1];
  const float* cls = (const float*)d_in[2];
  const float* f0 = (const float*)d_in[3];
  const float* f1 = (const float*)d_in[4];
  const float* f2 = (const float*)d_in[5];
  const float* f3 = (const float*)d_in[6];
  const float* f4 = (const float*)d_in[7];
  const int N = in_sizes[1] / 4;  // boxes flat = 1*N*4

  float* out_b = (float*)d_out;
  float* out_cls = out_b + K_TOP * 4;
  float* out_roi = out_cls + K_TOP * NCLS;

  uint32_t* ws = (uint32_t*)d_ws;
  uint32_t* ctl = ws;                   // 272 uints
  uint32_t* keys = ws + 512;            // N
  uint32_t* gtIdx = keys + N;           // K
  uint32_t* tieIdx = gtIdx + K_TOP;     // K
  uint32_t* perm = tieIdx + K_TOP;      // K
  int* lvl = (int*)(perm + K_TOP);      // K
  float* nb = (float*)(lvl + K_TOP);    // K*4

  const int nblk = (N + 255) / 256;

  k_init<<<1, 256, 0, stream>>>(ctl);
  k_scores<<<nblk, 256, 0, stream>>>(cls, keys, N);

  const int shifts[4] = {24, 16, 8, 0};
  for (int p = 0; p < 4; ++p) {
    uint32_t maskHi = (p == 0) ? 0u : (0xFFFFFFFFu << (shifts[p] + 8));
    k_hist<<<nblk, 256, 0, stream>>>(keys, ctl, N, shifts[p], maskHi);
    k_select<<<1, 256, 0, stream>>>(ctl, shifts[p]);
  }
  k_compact<<<nblk, 256, 0, stream>>>(keys, ctl, gtIdx, tieIdx, N);
  k_finalize<<<1, 512, 0, stream>>>(keys, gtIdx, tieIdx, ctl, boxes,
                                    image_shape, perm, lvl, nb, out_b);
  k_cls<<<(K_TOP * NCLS + 255) / 256, 256, 0, stream>>>(cls, perm, out_cls);

  dim3 grid(K_TOP, 14);
  k_roi<<<grid, 256, 0, stream>>>(f0, f1, f2, f3, f4, lvl, nb, out_roi);
}